// CTCLayer_v2_14078902796902
// MI455X (gfx1250) — compile-verified
//
#include <hip/hip_runtime.h>
#include <stdint.h>

// Problem dimensions (fixed by the reference's setup_inputs).
#define T_DIM 512
#define C_DIM 128
#define L_DIM 32
#define S_DIM 65          // 2L+1 extended states
#define BLANK 127         // C-1
#define CH    32          // timesteps per TDM tile (32 rows x 128 f32 = 16KB)
#define NCH   (T_DIM / CH)
#define EPSF  1e-7f
#define NEGF  -1e30f

typedef unsigned int v4u __attribute__((ext_vector_type(4)));
typedef int          v8i __attribute__((ext_vector_type(8)));
typedef int          v4i __attribute__((ext_vector_type(4)));

#if defined(__AMDGCN__) && __has_builtin(__builtin_amdgcn_tensor_load_to_lds) && \
    __has_builtin(__builtin_amdgcn_s_wait_tensorcnt)
#define USE_TDM 1
#else
#define USE_TDM 0
#endif

// 2-way log-sum-exp (for the final loss combine).
__device__ __forceinline__ float lae(float a, float b) {
  float m = fmaxf(a, b);
  return m + __logf(__expf(a - m) + __expf(b - m));
}

// Fused 2/3-way log-sum-exp: critical path = max3 -> exp -> add -> log,
// only 2 transcendentals deep, exps evaluate in parallel. use_c selects the
// skip-transition term via cndmask (branchless).
__device__ __forceinline__ float lse3(float a, float b, float c, bool use_c) {
  float m = fmaxf(a, b);              // v_max_f32
  m = use_c ? fmaxf(m, c) : m;        // v_max3/cndmask
  float s = __expf(a - m) + __expf(b - m);
  s += use_c ? __expf(c - m) : 0.0f;
  return m + __logf(s);
}

#if USE_TDM
// Issue a TDM load of a [rows x 128] f32 tile (contiguous in memory) to LDS.
// D# per CDNA5 ISA ch.8: group0 = {count/type/lds/global addr}, group1 = dims.
__device__ __forceinline__ void tdm_issue(const float* gsrc, unsigned lds_off, int rows) {
  unsigned long long ga = (unsigned long long)(uintptr_t)gsrc;
  v4u g0;
  g0.x = 1u;                                                // count=1, user D#
  g0.y = lds_off;                                           // lds_addr [63:32]
  g0.z = (unsigned)ga;                                      // global_addr [95:64]
  g0.w = (unsigned)((ga >> 32) & 0x1FFFFFFull) | (2u << 30);// addr[56:32] | type=2
  v8i g1;
  g1[0] = 2 << 16;          // data_size=2 (4 bytes); no multicast/pad/iterate
  g1[1] = C_DIM << 16;      // tensor_dim0 = 128 (low 16 bits at [63:48])
  g1[2] = rows << 16;       // tensor_dim0 hi=0 ; tensor_dim1 = rows (low 16)
  g1[3] = C_DIM << 16;      // tensor_dim1 hi=0 ; tile_dim0 = 128
  g1[4] = rows & 0xFFFF;    // tile_dim1 = rows ; tile_dim2 = 0 (2D tile)
  g1[5] = C_DIM;            // tensor_dim0_stride = 128 elements
  g1[6] = 0;
  g1[7] = 0;
  v4i z4 = {0, 0, 0, 0};
  v8i z8 = {0, 0, 0, 0, 0, 0, 0, 0};
  __builtin_amdgcn_tensor_load_to_lds(g0, g1, z4, z4, z8, 0);
}
#endif

__global__ __launch_bounds__(32) void ctc_loss_kernel(
    const float* __restrict__ y_pred,     // [B,T,C] softmax probs
    const int*   __restrict__ labels,     // [B,L]
    const int*   __restrict__ input_len,  // [B,1]
    const int*   __restrict__ label_len,  // [B,1]
    float*       __restrict__ out)        // [B,1]
{
  __shared__ float buf[2 * CH * C_DIM];   // double-buffered prob tiles (32KB)
  __shared__ int   labs[L_DIM];
  __shared__ float aLDS[96];              // alpha spill for final gather

  const int b    = blockIdx.x;
  const int lane = threadIdx.x;           // wave32: one wave per block
  const float* gbase = y_pred + (size_t)b * T_DIM * C_DIM;

  const unsigned lds0 = (unsigned)(uintptr_t)(&buf[0]);
  const unsigned lds1 = (unsigned)(uintptr_t)(&buf[CH * C_DIM]);

#if USE_TDM
  tdm_issue(gbase, lds0, CH);             // prefetch chunk 0 ASAP
#endif

  labs[lane] = labels[b * L_DIM + lane];
  __syncthreads();

  const int il   = input_len[b];
  const int ll   = label_len[b];
  const int last = 2 * ll;

  // Lane l owns extended states 3l, 3l+1, 3l+2 (states >= 65 are inert:
  // they start at NEG and transitions only move forward, never feeding back).
  const int sBase = 3 * lane;
  int  e[3];
  bool sk[3];
#pragma unroll
  for (int j = 0; j < 3; ++j) {
    int s = sBase + j;
    if (s < S_DIM && (s & 1)) {               // odd state: a label symbol
      e[j]  = labs[(s - 1) >> 1];
      sk[j] = (s >= 3) && (e[j] != labs[(s - 3) >> 1]);
    } else {                                  // even / padded: blank
      e[j]  = BLANK;
      sk[j] = false;
    }
  }

  float a0 = NEGF, a1 = NEGF, a2 = NEGF;
  float loss = 0.0f;

  for (int k = 0; k < NCH; ++k) {
#if USE_TDM
    if (k + 1 < NCH) {
      tdm_issue(gbase + (size_t)(k + 1) * CH * C_DIM,
                ((k + 1) & 1) ? lds1 : lds0, CH);
      __builtin_amdgcn_s_wait_tensorcnt(1);   // in-order TDM: chunk k is done
    } else {
      __builtin_amdgcn_s_wait_tensorcnt(0);
    }
    asm volatile("" ::: "memory");            // don't hoist LDS reads above wait
#else
    {
      const float* src = gbase + (size_t)k * CH * C_DIM;
      float* dst = &buf[(k & 1) * CH * C_DIM];
      for (int i = lane; i < CH * C_DIM; i += 32) dst[i] = src[i];
      __syncthreads();
    }
#endif
    const float* row = &buf[(k & 1) * C_DIM * CH];
    for (int r = 0; r < CH; ++r, row += C_DIM) {
      const int t = k * CH + r;
      // LDS gathers are independent of the alpha chain: issue first so the
      // log() of the gathered probs overlaps the recurrence.
      float lp0 = __logf(row[e[0]] + EPSF);
      float lp1 = __logf(row[e[1]] + EPSF);
      float lp2 = __logf(row[e[2]] + EPSF);
      if (t == 0) {
        a0 = a1 = a2 = NEGF;
        if (lane == 0) {
          a0 = __logf(row[BLANK] + EPSF);     // alpha0[0] = lp(blank)
          a1 = lp1;                           // alpha0[1] = lp(label 0)
        }
      } else {
        float am1 = __shfl_up(a2, 1);         // alpha[3l-1] from lane l-1
        float am2 = __shfl_up(a1, 1);         // alpha[3l-2] from lane l-1
        if (lane == 0) { am1 = NEGF; am2 = NEGF; }
        float n0 = lse3(a0, am1, am2, sk[0]);
        float n1 = lse3(a1, a0,  am1, sk[1]);
        float n2 = lse3(a2, a1,  a0,  sk[2]);
        a0 = n0 + lp0;
        a1 = n1 + lp1;
        a2 = n2 + lp2;
      }
      if (t == il - 1) {                      // wave-uniform branch
        aLDS[sBase]     = a0;
        aLDS[sBase + 1] = a1;
        aLDS[sBase + 2] = a2;
        __syncthreads();
        float la = aLDS[last];
        float lb = aLDS[last > 0 ? last - 1 : 0];
        loss = -lae(la, lb);
        __syncthreads();
      }
    }
#if !USE_TDM
    __syncthreads();
#endif
  }

  if (lane == 0) out[b] = loss;
}

extern "C" void kernel_launch(void* const* d_in, const int* in_sizes, int n_in,
                              void* d_out, int out_size, void* d_ws, size_t ws_size,
                              hipStream_t stream) {
  const float* y_pred    = (const float*)d_in[0];
  const int*   labels    = (const int*)d_in[1];
  const int*   input_len = (const int*)d_in[2];
  const int*   label_len = (const int*)d_in[3];
  float*       out       = (float*)d_out;
  const int B = in_sizes[2];                  // input_length is [B,1]
  ctc_loss_kernel<<<B, 32, 0, stream>>>(y_pred, labels, input_len, label_len, out);
}